// CustomLoss_17085379904346
// MI455X (gfx1250) — compile-verified
//
#include <hip/hip_runtime.h>
#include <math.h>

typedef __attribute__((ext_vector_type(2))) float v2f;
typedef __attribute__((ext_vector_type(8))) float v8f;

#define DDIM 128
#define KNBR 64
#define WAVES_PER_BLOCK 8
#define PAIRS_PER_WAVE 16

// One wave handles 16 (b,k) pairs (same b, 16 consecutive k).
// A-operand: diff^2 for 16 pairs x 4 dims per WMMA step (fp32, exact).
// B-operand: ones in column 0 -> C[m,0] accumulates ||P[i]-P[j_m]||^2.
__global__ __launch_bounds__(256)
void pair_norm_kernel(const float* __restrict__ P,
                      const int* __restrict__ iidx,
                      const int* __restrict__ jidx,
                      const float* __restrict__ S,
                      float* __restrict__ block_partials)
{
    __shared__ float lds[WAVES_PER_BLOCK];

    const int tid   = threadIdx.x;
    const int wave  = tid >> 5;
    const int lane  = tid & 31;
    const int gwave = blockIdx.x * WAVES_PER_BLOCK + wave;   // 0..16383
    const int b     = gwave >> 2;                            // 0..4095
    const int kbase = (gwave & 3) * PAIRS_PER_WAVE;          // 0,16,32,48
    const int m     = lane & 15;                             // pair within wave
    const int half  = lane >> 4;                             // K-half for A layout

    const int i = iidx[b];
    const int j = jidx[b * KNBR + kbase + m];
    const float* __restrict__ Pi = P + (size_t)i * DDIM;
    const float* __restrict__ Pj = P + (size_t)j * DDIM;

    v8f c = {0.f, 0.f, 0.f, 0.f, 0.f, 0.f, 0.f, 0.f};

    // B matrix (4x16): column 0 = 1.0 for every K row, all other columns 0.
    // Row-striped-across-lanes layout => both VGPRs are (N==0 ? 1 : 0).
    v2f bm;
    bm.x = (m == 0) ? 1.0f : 0.0f;
    bm.y = bm.x;

    // 32 steps of 4 dims: lane half=0 holds dims 4t+{0,1}, half=1 holds 4t+{2,3}.
#pragma unroll
    for (int t = 0; t < DDIM / 4; ++t) {
        const int d0 = 4 * t + 2 * half;
        v2f pi = *(const v2f*)(Pi + d0);   // broadcast across 16 lanes (L0 hit)
        v2f pj = *(const v2f*)(Pj + d0);   // gathered, 16B/row/step, streaming
        const float a0 = pi.x - pj.x;
        const float a1 = pi.y - pj.y;
        v2f A;
        A.x = a0 * a0;
        A.y = a1 * a1;
        // D = A(16x4,f32) x B(4x16,f32) + C ; C[m,0] += sum_d diff^2
        c = __builtin_amdgcn_wmma_f32_16x16x4_f32(
                /*neg_a=*/false, A, /*neg_b=*/false, bm,
                /*c_mod=*/(short)0, c, /*reuse_a=*/false, /*reuse_b=*/false);
    }

    // C layout: lane 0 holds C[m=0..7, 0] in c[0..7]; lane 16 holds C[m=8..15, 0].
    float partial = 0.0f;
    if (m == 0) {
        const int mb = half * 8;
        const float* __restrict__ Sp = S + b * KNBR + kbase + mb;
#pragma unroll
        for (int r = 0; r < 8; ++r)
            partial += sqrtf(c[r]) * Sp[r];
    }
    // combine lane0 + lane16 (wave32)
    partial += __shfl_xor(partial, 16);

    if (lane == 0) lds[wave] = partial;
    __syncthreads();
    if (tid == 0) {
        float s = 0.0f;
#pragma unroll
        for (int w = 0; w < WAVES_PER_BLOCK; ++w) s += lds[w];
        block_partials[blockIdx.x] = s;   // fixed order -> deterministic
    }
}

// Single-block deterministic final reduction of the 2048 block partials.
__global__ __launch_bounds__(256)
void final_reduce_kernel(const float* __restrict__ partials, int n,
                         float* __restrict__ out)
{
    __shared__ float lds[256];
    float s = 0.0f;
    for (int idx = threadIdx.x; idx < n; idx += 256)
        s += partials[idx];               // fixed per-thread order
    lds[threadIdx.x] = s;
    __syncthreads();
#pragma unroll
    for (int off = 128; off > 0; off >>= 1) {
        if (threadIdx.x < off) lds[threadIdx.x] += lds[threadIdx.x + off];
        __syncthreads();
    }
    if (threadIdx.x == 0) out[0] = lds[0];
}

extern "C" void kernel_launch(void* const* d_in, const int* in_sizes, int n_in,
                              void* d_out, int out_size, void* d_ws, size_t ws_size,
                              hipStream_t stream)
{
    const float* P    = (const float*)d_in[0];
    const int*   iidx = (const int*)d_in[1];
    const int*   jidx = (const int*)d_in[2];
    const float* S    = (const float*)d_in[3];

    const int B       = in_sizes[1];                     // 4096
    const int nPairs  = B * KNBR;                        // 262144
    const int nWaves  = nPairs / PAIRS_PER_WAVE;         // 16384
    const int nBlocks = nWaves / WAVES_PER_BLOCK;        // 2048

    float* partials = (float*)d_ws;                      // 2048 floats

    pair_norm_kernel<<<nBlocks, 256, 0, stream>>>(P, iidx, jidx, S, partials);
    final_reduce_kernel<<<1, 256, 0, stream>>>(partials, nBlocks, (float*)d_out);
}